// PrivateGraphSAGE_14121852470182
// MI455X (gfx1250) — compile-verified
//
#include <hip/hip_runtime.h>
#include <hip/hip_bf16.h>
#include <math.h>

#define N_NODES 50000
#define N_EDGES 600000
#define D 128
#define LDSTRIDE 130  // padded row stride (floats) to avoid LDS bank conflicts

typedef __attribute__((ext_vector_type(2))) float v2f;
typedef __attribute__((ext_vector_type(8))) float v8f;

// ---------------------------------------------------------------------------
// Kernel 1: per-row L2 clip + combine with DP noise.
//   xc[row]  = x[row] / max(||x[row]||, 1)
//   acc[row] = xc[row] + noise[row]          (accumulator init = self + noise)
// One wave32 per row: 128 floats = 4 per lane (float4).
// Safe to run with x == xc (in-place): each lane reads before it writes.
// ---------------------------------------------------------------------------
__global__ __launch_bounds__(256) void clip_combine_kernel(
    const float* x, const float* __restrict__ noise,
    float* xc, float* __restrict__ acc) {
  const int row  = (blockIdx.x * blockDim.x + threadIdx.x) >> 5;
  const int lane = threadIdx.x & 31;
  if (row >= N_NODES) return;

  float4 v = ((const float4*)(x + (size_t)row * D))[lane];
  float ss = v.x * v.x + v.y * v.y + v.z * v.z + v.w * v.w;
#pragma unroll
  for (int off = 16; off > 0; off >>= 1)
    ss += __shfl_xor(ss, off, 32);          // wave32 butterfly reduce
  const float n = sqrtf(ss);
  const float s = (n > 1.0f) ? (1.0f / n) : 1.0f;
  v.x *= s; v.y *= s; v.z *= s; v.w *= s;
  ((float4*)(xc + (size_t)row * D))[lane] = v;

  const float4 nz = ((const float4*)(noise + (size_t)row * D))[lane];
  float4 a; a.x = v.x + nz.x; a.y = v.y + nz.y; a.z = v.z + nz.z; a.w = v.w + nz.w;
  ((float4*)(acc + (size_t)row * D))[lane] = a;
}

// ---------------------------------------------------------------------------
// Kernel 2: edge scatter-add. One wave32 per edge: lane loads a float4 of the
// source row and issues 4 native fp32 atomics into the destination row.
// Both xc (gather src) and acc (atomic dst) are 25.6 MB -> L2-resident
// (192 MB L2), so the 77M atomics/layer resolve on-chip.
// ---------------------------------------------------------------------------
__global__ __launch_bounds__(256) void edge_scatter_kernel(
    const float* __restrict__ xc, const int* __restrict__ ei,
    float* __restrict__ acc) {
  const int e    = (blockIdx.x * blockDim.x + threadIdx.x) >> 5;
  const int lane = threadIdx.x & 31;
  if (e >= N_EDGES) return;
  const int src = ei[e];             // edge_index[0, e]
  const int dst = ei[N_EDGES + e];   // edge_index[1, e]
  const float4 v = ((const float4*)(xc + (size_t)src * D))[lane];
  float* out = acc + (size_t)dst * D + lane * 4;
  unsafeAtomicAdd(out + 0, v.x);
  unsafeAtomicAdd(out + 1, v.y);
  unsafeAtomicAdd(out + 2, v.z);
  unsafeAtomicAdd(out + 3, v.w);
}

// ---------------------------------------------------------------------------
// Kernel 3: out = A @ W^T + bias, optional fused SELU.  fp32 WMMA path.
//   A: [N_NODES, D] row-major, W: [D, D] row-major (out[m,n] = sum_k A[m,k]*W[n,k])
// Block = 256 threads = 8 waves; one 16-row M-tile per block.
//   - stage W (64 KB) + A-tile (16x128) in LDS, padded stride to avoid
//     bank conflicts
//   - wave w computes N-tile [16w, 16w+16): 32x v_wmma_f32_16x16x4_f32
// ISA fp32 16x16x4 operand layout (CDNA5 ISA):
//   A frag: lane(0-15)=row M, v0/v1 = K0,K1 ; lanes 16-31 = K2,K3
//   B frag: mirrored (K x N), lane&15 = column N, half selects K pair
//   C/D   : VGPR r -> row r + 8*(lane>=16), col = lane&15
// ---------------------------------------------------------------------------
template <bool APPLY_SELU>
__global__ __launch_bounds__(256) void gemm_bias_wmma_kernel(
    const float* __restrict__ A, const float* __restrict__ W,
    const float* __restrict__ bias, float* __restrict__ out) {
  __shared__ float Ws[D * LDSTRIDE];   // ~65 KB
  __shared__ float As[16 * LDSTRIDE];  // ~8.3 KB

  const int tid = threadIdx.x;
  const int m0  = blockIdx.x * 16;

  // Stage W: 16384 floats, coalesced float4, scattered into padded rows.
#pragma unroll
  for (int i = 0; i < (D * D / 4) / 256; ++i) {
    const int idx = (tid + i * 256) * 4;      // flat float index in W
    const int r = idx >> 7, c = idx & (D - 1);
    const float4 w4 = *(const float4*)(W + idx);
    *(float4*)(Ws + r * LDSTRIDE + c) = w4;   // c % 4 == 0, but stride 130 -> odd rows only 8B aligned
  }
  // Stage A-tile: 2048 floats.
#pragma unroll
  for (int i = 0; i < (16 * D / 4) / 256; ++i) {
    const int idx = (tid + i * 256) * 4;
    const int r = idx >> 7, c = idx & (D - 1);
    const float4 a4 = *(const float4*)(A + (size_t)m0 * D + idx);
    *(float4*)(As + r * LDSTRIDE + c) = a4;
  }
  __syncthreads();

  const int lane = tid & 31;
  const int half = lane >> 4;   // selects K pair within a k0 step
  const int l    = lane & 15;   // row (A) / column (B) within the tile
  const int n0   = (tid >> 5) * 16;

  v8f acc = {};
#pragma unroll
  for (int k0 = 0; k0 < D; k0 += 4) {
    const int kk = k0 + 2 * half;
    const v2f a = *(const v2f*)(As + l * LDSTRIDE + kk);          // A[m0+l][kk..kk+1]
    const v2f b = *(const v2f*)(Ws + (n0 + l) * LDSTRIDE + kk);   // W[n0+l][kk..kk+1] = B[kk][n0+l]
    acc = __builtin_amdgcn_wmma_f32_16x16x4_f32(
        /*neg_a=*/false, a, /*neg_b=*/false, b,
        /*c_mod=*/(short)0, acc, /*reuse_a=*/false, /*reuse_b=*/false);
  }

  const float bv = bias[n0 + l];
#pragma unroll
  for (int r = 0; r < 8; ++r) {
    float v = acc[r] + bv;
    if (APPLY_SELU) {
      const float lam   = 1.0507009873554805f;
      const float alpha = 1.6732632423543772f;
      v = (v > 0.0f) ? lam * v : lam * alpha * expm1f(v);
    }
    out[(size_t)(m0 + r + 8 * half) * D + n0 + l] = v;
  }
}

// ---------------------------------------------------------------------------
// Orchestration: clip1 -> scatter1 -> WMMA GEMM1+SELU -> clip2 -> scatter2
//                -> WMMA GEMM2 -> d_out.  All on `stream`, graph-capture safe.
// Workspace: 2 x (N_NODES*D) f32 buffers = 51.2 MB.
// ---------------------------------------------------------------------------
extern "C" void kernel_launch(void* const* d_in, const int* in_sizes, int n_in,
                              void* d_out, int out_size, void* d_ws, size_t ws_size,
                              hipStream_t stream) {
  const float* x      = (const float*)d_in[0];
  const int*   ei     = (const int*)d_in[1];   // [2, N_EDGES] row-major
  const float* W1     = (const float*)d_in[2];
  const float* b1     = (const float*)d_in[3];
  const float* W2     = (const float*)d_in[4];
  const float* b2     = (const float*)d_in[5];
  const float* noise1 = (const float*)d_in[6];
  const float* noise2 = (const float*)d_in[7];
  float* out = (float*)d_out;

  float* buf0 = (float*)d_ws;                       // xc / h / hc
  float* buf1 = buf0 + (size_t)N_NODES * D;         // aggregation accumulator

  const int clipBlocks    = (N_NODES * 32 + 255) / 256;
  const int scatterBlocks = (N_EDGES * 32 + 255) / 256;
  const int gemmBlocks    = N_NODES / 16;           // 3125

  // ---- Layer 1 ----
  clip_combine_kernel<<<clipBlocks, 256, 0, stream>>>(x, noise1, buf0, buf1);
  edge_scatter_kernel<<<scatterBlocks, 256, 0, stream>>>(buf0, ei, buf1);
  gemm_bias_wmma_kernel<true><<<gemmBlocks, 256, 0, stream>>>(buf1, W1, b1, buf0);

  // ---- Layer 2 ----
  clip_combine_kernel<<<clipBlocks, 256, 0, stream>>>(buf0, noise2, buf0, buf1);
  edge_scatter_kernel<<<scatterBlocks, 256, 0, stream>>>(buf0, ei, buf1);
  gemm_bias_wmma_kernel<false><<<gemmBlocks, 256, 0, stream>>>(buf1, W2, b2, out);
}